// SoftPolygonCUDA_39092792329016
// MI455X (gfx1250) — compile-verified
//
#include <hip/hip_runtime.h>
#include <math.h>

typedef __attribute__((ext_vector_type(2))) float v2f;
typedef __attribute__((ext_vector_type(8))) float v8f;

#define NV  64         // vertices / edges per polygon
#define WPX 64         // width
#define HPX 64         // height

// One wave32 handles a 16-pixel row segment of one batch image.
// Edges ride the M dimension (A matrix, 4 pre-packed variants in LDS),
// pixels ride the N dimension (B matrix, built in registers).
// Per 16-edge tile, 4x V_WMMA_F32_16X16X4_F32 give, per (edge m, pixel n):
//   S1 = pa.d          A1 row = [dx, dy, -(a.d), 0]
//   U  = pa.d / dd     Au row = A1 / dd
//   Q  = |p-a|^2       Aq row = [-2ax, -2ay, |a|^2, 1]
//   QB = |p-b|^2       Ab row = [-2bx, -2by, |b|^2, 1]
// Clamped segment distance: u<=0 -> Q ; u>=1 -> QB ; else Q - S1*U.
// Min over edges = min across the 8 C registers (VALU) + one shfl_xor(16).
__global__ __launch_bounds__(256) void soft_polygon_kernel(
    const float* __restrict__ verts,   // (B, 64, 2)
    float* __restrict__ out)           // (B, 64, 64)
{
    __shared__ float2 sA1[4][32];   // [edge tile][lane] : pa.d operand
    __shared__ float2 sAu[4][32];   // [edge tile][lane] : (pa.d)/dd operand
    __shared__ float2 sAq[4][32];   // [edge tile][lane] : |p-a|^2 operand
    __shared__ float2 sAb[4][32];   // [edge tile][lane] : |p-b|^2 operand
    __shared__ float4 sE[NV];       // (ax, ay, bx, by)
    __shared__ float4 sX4[8][16];   // per-wave xeff[64] for crossing test

    const int batch = blockIdx.x >> 5;             // 32 blocks per batch
    const int tid   = threadIdx.x;

    // ---- Stage per-edge data, pre-packed in WMMA A-operand lane layout ----
    if (tid < NV) {
        const float* vb = verts + batch * (NV * 2);
        float ax = vb[2 * tid + 0];
        float ay = vb[2 * tid + 1];
        int t1 = (tid + 1) & (NV - 1);
        float bx = vb[2 * t1 + 0];
        float by = vb[2 * t1 + 1];
        float dx = bx - ax, dy = by - ay;
        float dd = fmaxf(dx * dx + dy * dy, 1e-12f);
        float iv = 1.0f / dd;
        float ad = ax * dx + ay * dy;
        int e = tid >> 4, slot = tid & 15;
        // A 16x4 layout: lanes 0-15 hold K0,K1 ; lanes 16-31 hold K2,K3
        sA1[e][slot]      = make_float2(dx, dy);
        sA1[e][slot + 16] = make_float2(-ad, 0.0f);
        sAu[e][slot]      = make_float2(dx * iv, dy * iv);
        sAu[e][slot + 16] = make_float2(-ad * iv, 0.0f);
        sAq[e][slot]      = make_float2(-2.0f * ax, -2.0f * ay);
        sAq[e][slot + 16] = make_float2(ax * ax + ay * ay, 1.0f);
        sAb[e][slot]      = make_float2(-2.0f * bx, -2.0f * by);
        sAb[e][slot + 16] = make_float2(bx * bx + by * by, 1.0f);
        sE[tid] = make_float4(ax, ay, bx, by);
    }
    __syncthreads();

    const int wave = tid >> 5;
    const int lane = tid & 31;
    const int nlo  = lane & 15;
    const bool hi  = lane >= 16;

    // Pixel tile: 16 contiguous pixels of one image row.
    const int tile = ((blockIdx.x & 31) << 3) + wave;   // 0..255
    const int ty = tile >> 2;                           // row 0..63
    const int tx = (tile & 3) << 4;                     // col base
    const float py  = (float)ty;
    const float pxn = (float)(tx + nlo);                // px for N = nlo

    // B (4x16 fp32, pixels): col_n = [px, py, 1, px^2+py^2]
    v2f Bp;
    Bp.x = hi ? 1.0f : pxn;
    Bp.y = hi ? fmaf(pxn, pxn, py * py) : py;

    // ---- Per-wave crossing precompute: xeff per edge (py is wave-uniform) ----
    // crossing iff (ay>y)!=(by>y) && x < x_int  ->  x < xeff
    float* sX = (float*)&sX4[wave][0];
#pragma unroll
    for (int h = 0; h < 2; ++h) {
        int n = lane + 32 * h;
        float4 E = sE[n];
        bool cond = (E.y > py) != (E.w > py);
        float xint = fmaf(E.z - E.x, (py - E.y) / (E.w - E.y), E.x);
        sX[n] = cond ? xint : -1.0e30f;
    }

    // ---- Main loop: 4 edge tiles x 4 WMMAs ----
    float macc = 3.4e38f;
#pragma unroll
    for (int e = 0; e < 4; ++e) {
        v2f A1 = *(const v2f*)&sA1[e][lane];
        v2f Au = *(const v2f*)&sAu[e][lane];
        v2f Aq = *(const v2f*)&sAq[e][lane];
        v2f Ab = *(const v2f*)&sAb[e][lane];

        v8f S1 = {}, U = {}, Q = {}, QB = {};
        S1 = __builtin_amdgcn_wmma_f32_16x16x4_f32(
                false, A1, false, Bp, (short)0, S1, false, false);
        U  = __builtin_amdgcn_wmma_f32_16x16x4_f32(
                false, Au, false, Bp, (short)0, U, false, false);
        Q  = __builtin_amdgcn_wmma_f32_16x16x4_f32(
                false, Aq, false, Bp, (short)0, Q, false, false);
        QB = __builtin_amdgcn_wmma_f32_16x16x4_f32(
                false, Ab, false, Bp, (short)0, QB, false, false);

#pragma unroll
        for (int r = 0; r < 8; ++r) {
            float u  = U[r];
            float d2 = fmaf(-S1[r], u, Q[r]);       // interior: q - s1*u
            d2 = (u <= 0.0f) ? Q[r]  : d2;          // before a: |p-a|^2
            d2 = (u >= 1.0f) ? QB[r] : d2;          // past b:   |p-b|^2
            macc = fminf(macc, d2);
        }
    }

    // ---- Crossing count: lanes 0-15 scan edges 0-31, lanes 16-31 scan 32-63 ----
    int c = 0;
    {
        const float4* xv = &sX4[wave][hi ? 8 : 0];
        const float x = pxn;
#pragma unroll
        for (int k = 0; k < 8; ++k) {
            float4 xe = xv[k];
            c += (int)(x < xe.x) + (int)(x < xe.y) +
                 (int)(x < xe.z) + (int)(x < xe.w);
        }
    }

    // ---- Merge the two lane halves (each covers half the edges) ----
    float m2 = fminf(macc, __shfl_xor(macc, 16, 32));
    int   ct = c + __shfl_xor(c, 16, 32);

    float z = ((ct & 1) ? 10.0f : -10.0f) * m2;     // sign * min_d2 / 0.1
    if (lane < 16) {
        out[batch * (HPX * WPX) + ty * WPX + tx + nlo] =
            1.0f / (1.0f + expf(-z));
    }
}

extern "C" void kernel_launch(void* const* d_in, const int* in_sizes, int n_in,
                              void* d_out, int out_size, void* d_ws, size_t ws_size,
                              hipStream_t stream) {
    const float* verts = (const float*)d_in[0];
    float* out = (float*)d_out;
    const int B = in_sizes[0] / (NV * 2);   // 128
    dim3 grid(B * 32);                      // 32 blocks per batch (8 waves each)
    dim3 block(256);
    soft_polygon_kernel<<<grid, block, 0, stream>>>(verts, out);
}